// FactorizedAttention_66451734004227
// MI455X (gfx1250) — compile-verified
//
#include <hip/hip_runtime.h>
#include <math.h>

// Problem constants (from reference)
constexpr int NB = 32;    // batch
constexpr int TT = 50;    // time
constexpr int LQn = 64;   // query len
constexpr int LKn = 64;   // key len
constexpr int EE = 128;   // embed
constexpr int HH = 8;     // heads
constexpr int DD = 16;    // head dim

typedef __attribute__((ext_vector_type(2))) float v2f;
typedef __attribute__((ext_vector_type(8))) float v8f;

__device__ __forceinline__ v8f wmma_f32_k4(v2f a, v2f b, v8f c) {
  // D = A(16x4 f32) * B(4x16 f32) + C(16x16 f32)
  return __builtin_amdgcn_wmma_f32_16x16x4_f32(
      /*neg_a=*/false, a, /*neg_b=*/false, b,
      /*c_mod=*/(short)0, c, /*reuse_a=*/false, /*reuse_b=*/false);
}

// ---------------------------------------------------------------------------
// Y[M x 128] = X[M x 128] @ W^T[128 x 128] + bias    (W row-major [out][in])
// Block: 256 threads = 8 waves. Block owns 16 rows; wave w owns cols 16w..16w+15.
// ---------------------------------------------------------------------------
__global__ __launch_bounds__(256) void proj_gemm(
    const float* __restrict__ X, const float* __restrict__ W,
    const float* __restrict__ bias, float* __restrict__ Y) {
  const int lane = threadIdx.x & 31;
  const int wave = threadIdx.x >> 5;
  const int mn   = lane & 15;          // A row (m) / B col (n)
  const int kb   = (lane >> 4) << 1;   // K sub-chunk base: 0 or 2
  const int row0 = blockIdx.x << 4;
  const int col0 = wave << 4;

  const float* xr = X + (size_t)(row0 + mn) * EE;  // A: X[row0+m][e0+kb..+1]
  const float* wr = W + (size_t)(col0 + mn) * EE;  // B: W[col0+n][e0+kb..+1]

  v8f acc = {0.f, 0.f, 0.f, 0.f, 0.f, 0.f, 0.f, 0.f};
#pragma unroll
  for (int e0 = 0; e0 < EE; e0 += 4) {
    v2f a = *(const v2f*)(xr + e0 + kb);
    v2f b = *(const v2f*)(wr + e0 + kb);
    acc = wmma_f32_k4(a, b, acc);
  }

  const float bv = bias[col0 + mn];
  const int rbase = row0 + ((lane >> 4) << 3);     // row = rbase + r
  float* yb = Y + (size_t)rbase * EE + col0 + mn;
#pragma unroll
  for (int r = 0; r < 8; ++r) yb[(size_t)r * EE] = acc[r] + bv;
}

// ---------------------------------------------------------------------------
// Attention per (b,t,h). Block: 128 threads = 4 waves; wave w owns q rows
// [16w, 16w+16). scores(64x64) -> mask -> softmax -> attn_out + LDS -> P@V.
// ---------------------------------------------------------------------------
__global__ __launch_bounds__(128) void attn_kernel(
    const float* __restrict__ q, const float* __restrict__ k,
    const float* __restrict__ v, const unsigned char* __restrict__ mask,
    float* __restrict__ attn_out, float* __restrict__ ctx) {
  __shared__ float lds[4][16][LKn];   // 16 KB: per-wave prob tile

  const int lane  = threadIdx.x & 31;
  const int wave  = threadIdx.x >> 5;
  const int mn    = lane & 15;
  const int hi    = lane >> 4;     // half select
  const int kb    = hi << 1;       // K sub-chunk base for A/B frags
  const int rhalf = hi << 3;       // row offset for C/D frags

  const int h  = blockIdx.x % HH;
  const int bt = blockIdx.x / HH;  // b*T + t
  const int l0 = wave << 4;

  const float* qbase = q + (size_t)bt * LQn * EE + h * DD;
  const float* kbase = k + (size_t)bt * LKn * EE + h * DD;
  const float* vbase = v + (size_t)bt * LKn * EE + h * DD;
  const unsigned char* mrow = mask + (size_t)bt * LKn;

  // ---- scores = Q @ K^T (contraction over d=16, 4 chunks of 4) ----
  v2f afrag[4];
#pragma unroll
  for (int c = 0; c < 4; ++c) {
    const float* p = qbase + (size_t)(l0 + mn) * EE + 4 * c + kb;
    afrag[c].x = p[0];
    afrag[c].y = p[1];
  }
  v8f s[4];
#pragma unroll
  for (int j = 0; j < 4; ++j) {  // key col tiles: cols 16j..16j+15
    v8f acc = {0.f, 0.f, 0.f, 0.f, 0.f, 0.f, 0.f, 0.f};
#pragma unroll
    for (int c = 0; c < 4; ++c) {
      const float* p = kbase + (size_t)(16 * j + mn) * EE + 4 * c + kb;
      v2f bf;
      bf.x = p[0];
      bf.y = p[1];
      acc = wmma_f32_k4(afrag[c], bf, acc);
    }
    s[j] = acc;
  }

  bool msk[4];
#pragma unroll
  for (int j = 0; j < 4; ++j) msk[j] = (mrow[16 * j + mn] != 0);

  const float scale = 0.25f;  // 1/sqrt(16)

  // ---- row softmax; row = l0 + rhalf + r spans 16 lanes x 4 fragments ----
#pragma unroll
  for (int r = 0; r < 8; ++r) {
    float x[4];
    float rowmax = -INFINITY;
#pragma unroll
    for (int j = 0; j < 4; ++j) {
      x[j] = msk[j] ? -INFINITY : s[j][r] * scale;
      rowmax = fmaxf(rowmax, x[j]);
    }
#pragma unroll
    for (int m = 1; m < 16; m <<= 1)
      rowmax = fmaxf(rowmax, __shfl_xor(rowmax, m, 32));
    float sum = 0.f;
#pragma unroll
    for (int j = 0; j < 4; ++j) {
      float p = (rowmax == -INFINITY) ? 0.f : __expf(x[j] - rowmax);
      x[j] = p;
      sum += p;
    }
#pragma unroll
    for (int m = 1; m < 16; m <<= 1) sum += __shfl_xor(sum, m, 32);
    const float inv = (sum > 0.f) ? (1.f / sum) : 0.f;  // nan_to_num semantics
#pragma unroll
    for (int j = 0; j < 4; ++j) s[j][r] = x[j] * inv;
  }

  // ---- write attn probs: global layout (h, b, t, l, k) + LDS for PV ----
  float* abase = attn_out + ((size_t)h * NB * TT + bt) * (size_t)(LQn * LKn);
#pragma unroll
  for (int r = 0; r < 8; ++r) {
    const int l = l0 + rhalf + r;
#pragma unroll
    for (int j = 0; j < 4; ++j) {
      const float pv = s[j][r];
      abase[(size_t)l * LKn + 16 * j + mn] = pv;
      lds[wave][rhalf + r][16 * j + mn] = pv;
    }
  }
  __syncthreads();

  // ---- context = P(16x64) @ V(64x16): 16 chunks of K=4 ----
  v8f o = {0.f, 0.f, 0.f, 0.f, 0.f, 0.f, 0.f, 0.f};
#pragma unroll
  for (int c = 0; c < 16; ++c) {
    const int kk0 = 4 * c + kb;
    v2f a;
    a.x = lds[wave][mn][kk0];
    a.y = lds[wave][mn][kk0 + 1];
    v2f bf;
    bf.x = vbase[(size_t)kk0 * EE + mn];
    bf.y = vbase[(size_t)(kk0 + 1) * EE + mn];
    o = wmma_f32_k4(a, bf, o);
  }
  float* cb = ctx + ((size_t)bt * LQn + l0 + rhalf) * EE + h * DD + mn;
#pragma unroll
  for (int r = 0; r < 8; ++r) cb[(size_t)r * EE] = o[r];
}

// ---------------------------------------------------------------------------
extern "C" void kernel_launch(void* const* d_in, const int* in_sizes, int n_in,
                              void* d_out, int out_size, void* d_ws,
                              size_t ws_size, hipStream_t stream) {
  (void)in_sizes; (void)n_in; (void)out_size; (void)ws_size;

  const float* query = (const float*)d_in[0];
  const float* key   = (const float*)d_in[1];
  const float* value = (const float*)d_in[2];
  const unsigned char* kpm = (const unsigned char*)d_in[3];  // bool mask
  const float* Wq = (const float*)d_in[4];
  const float* bq = (const float*)d_in[5];
  const float* Wk = (const float*)d_in[6];
  const float* bk = (const float*)d_in[7];
  const float* Wv = (const float*)d_in[8];
  const float* bv = (const float*)d_in[9];
  const float* Wo = (const float*)d_in[10];
  const float* bo = (const float*)d_in[11];

  float* out = (float*)d_out;
  const size_t OUT_MAIN = (size_t)NB * TT * LQn * EE;  // 13,107,200 floats
  float* attn_out = out + OUT_MAIN;                    // (H,N,T,LQ,LK)

  float* ws  = (float*)d_ws;
  const size_t QKV = OUT_MAIN;  // per-tensor projected size
  float* qp  = ws;
  float* kp  = ws + QKV;
  float* vp  = ws + 2 * QKV;
  float* ctx = ws + 3 * QKV;

  const int Mtiles = (NB * TT * LQn) / 16;  // 6400 row tiles (LK==LQ)

  proj_gemm<<<dim3(Mtiles), dim3(256), 0, stream>>>(query, Wq, bq, qp);
  proj_gemm<<<dim3(Mtiles), dim3(256), 0, stream>>>(key,   Wk, bk, kp);
  proj_gemm<<<dim3(Mtiles), dim3(256), 0, stream>>>(value, Wv, bv, vp);
  attn_kernel<<<dim3(NB * TT * HH), dim3(128), 0, stream>>>(qp, kp, vp, kpm,
                                                            attn_out, ctx);
  proj_gemm<<<dim3(Mtiles), dim3(256), 0, stream>>>(ctx, Wo, bo, out);
}